// Simulator_75376676044885
// MI455X (gfx1250) — compile-verified
//
#include <hip/hip_runtime.h>
#include <math.h>

typedef __attribute__((ext_vector_type(2))) float v2f;
typedef __attribute__((ext_vector_type(8))) float v8f;

#define TIMESTEP 0.02f
#define NBINS 140           // N_BINS_POT
#define PAIR_CEN_W 138      // dist_bin_centres row width
#define ANG_POT_W 140
#define DIH_POT_W 142
#define MAX_STEPS 8
static const double PI_D = 3.14159265358979323846;

struct F3 { float x, y, z; };
__device__ __forceinline__ F3 mk3(float x, float y, float z) { F3 r; r.x = x; r.y = y; r.z = z; return r; }
__device__ __forceinline__ F3 sub3(F3 a, F3 b) { return mk3(a.x - b.x, a.y - b.y, a.z - b.z); }
__device__ __forceinline__ F3 add3(F3 a, F3 b) { return mk3(a.x + b.x, a.y + b.y, a.z + b.z); }
__device__ __forceinline__ F3 neg3(F3 a) { return mk3(-a.x, -a.y, -a.z); }
__device__ __forceinline__ F3 scl3(F3 a, float s) { return mk3(a.x * s, a.y * s, a.z * s); }
__device__ __forceinline__ float dot3(F3 a, F3 b) { return a.x * b.x + a.y * b.y + a.z * b.z; }
__device__ __forceinline__ F3 cross3(F3 a, F3 b) {
    return mk3(a.y * b.z - a.z * b.y, a.z * b.x - a.x * b.z, a.x * b.y - a.y * b.x);
}
__device__ __forceinline__ float safe_norm3(F3 a) {
    float sq = dot3(a, a);
    return sq > 0.f ? sqrtf(sq) : 0.f;
}
__device__ __forceinline__ F3 normalize3(F3 a) {
    float n = fmaxf(safe_norm3(a), 1e-12f);
    return scl3(a, 1.f / n);
}
__device__ __forceinline__ F3 loadc(const float* c, int idx) {
    return mk3(c[3 * idx], c[3 * idx + 1], c[3 * idx + 2]);
}

// ---------------- init: copy coords/vels into workspace, zero acc_last ----------------
__global__ void k_init(const float* __restrict__ cin, const float* __restrict__ vin,
                       float* wc, float* wv, float* wal, int total) {
    int i = blockIdx.x * blockDim.x + threadIdx.x;
    if (i < total) { wc[i] = cin[i]; wv[i] = vin[i]; wal[i] = 0.f; }
}

// ---------------- position half-kick ----------------
__global__ void k_pos(float* wc, const float* __restrict__ wv, const float* __restrict__ wal,
                      int total, int step, const int* __restrict__ nsteps) {
    if (step >= *nsteps) return;
    int i = blockIdx.x * blockDim.x + threadIdx.x;
    if (i < total) wc[i] += wv[i] * TIMESTEP + 0.5f * wal[i] * TIMESTEP * TIMESTEP;
}

// ---------------- squared norms of each atom position ----------------
__global__ void k_norms(const float* __restrict__ wc, float* wn, int bn,
                        int step, const int* __restrict__ nsteps) {
    if (step >= *nsteps) return;
    int i = blockIdx.x * blockDim.x + threadIdx.x;
    if (i < bn) {
        float x = wc[3 * i], y = wc[3 * i + 1], z = wc[3 * i + 2];
        wn[i] = x * x + y * y + z * z;
    }
}

// ---------------- pairwise distance matrix via WMMA Gram tiles ----------------
// One wave per 16x16 output tile. dist(i,j) = sqrt(max(ni + nj - 2*ci.cj, 0))
__global__ __launch_bounds__(32) void k_gram(const float* __restrict__ wc,
                                             const float* __restrict__ wn,
                                             float* __restrict__ wdist, int n,
                                             int step, const int* __restrict__ nsteps) {
    if (step >= *nsteps) return;
    int bi = blockIdx.z;
    int iBase = blockIdx.y * 16;
    int jBase = blockIdx.x * 16;
    const float* c = wc + (size_t)bi * n * 3;
    const float* nm = wn + (size_t)bi * n;
    float* dmat = wdist + (size_t)bi * n * n;

    int lane = threadIdx.x;
    int m = lane & 15;
    bool hi = lane >= 16;         // upper half-wave holds K = 2,3

    // A: 16x4 (M x K), A[m][0..2] = coords, A[m][3] = 0
    int arow = iBase + m;
    v2f a; a.x = 0.f; a.y = 0.f;
    if (arow < n) {
        const float* cr = c + 3 * arow;
        a.x = hi ? cr[2] : cr[0];
        a.y = hi ? 0.f   : cr[1];
    }
    // B: 4x16 (K x N), B[k][nn] = coords[jBase+nn][k]
    int bcol = jBase + m;
    v2f bv; bv.x = 0.f; bv.y = 0.f;
    if (bcol < n) {
        const float* cr = c + 3 * bcol;
        bv.x = hi ? cr[2] : cr[0];
        bv.y = hi ? 0.f   : cr[1];
    }

    v8f acc = {};
    acc = __builtin_amdgcn_wmma_f32_16x16x4_f32(false, a, false, bv, (short)0, acc,
                                                false, false);

    int j = jBase + m;                       // column N = lane & 15
    float nj = (j < n) ? nm[j] : 0.f;
#pragma unroll
    for (int r = 0; r < 8; ++r) {
        int Mi = iBase + r + (hi ? 8 : 0);   // C/D layout: VGPR r -> M = r (+8 upper half)
        if (Mi < n && j < n) {
            float d2 = nm[Mi] + nj - 2.f * acc[r];
            dmat[(size_t)Mi * n + j] = d2 > 0.f ? sqrtf(d2) : 0.f;
        }
    }
}

// ---------------- pairwise forces: one block per receiving atom j ----------------
__global__ __launch_bounds__(256) void k_pair(const float* __restrict__ wc,
                                              const float* __restrict__ wdist,
                                              const float* __restrict__ masses,
                                              const float* __restrict__ ff_dist,
                                              const float* __restrict__ bin_cen,
                                              const int* __restrict__ inters_flat,
                                              float* __restrict__ wacc, int n,
                                              int step, const int* __restrict__ nsteps) {
    if (step >= *nsteps) return;
    int bi = blockIdx.y;
    int j = blockIdx.x;
    const float* c = wc + (size_t)bi * n * 3;
    const float* dmat = wdist + (size_t)bi * n * n;
    float cjx = c[3 * j], cjy = c[3 * j + 1], cjz = c[3 * j + 2];

    float fx = 0.f, fy = 0.f, fz = 0.f;
    for (int i = threadIdx.x; i < n; i += blockDim.x) {
        size_t flat = (size_t)i * n + j;
        int type = inters_flat[flat];
        float d = dmat[flat];
        const float* cen = bin_cen + (size_t)type * PAIR_CEN_W;
        int best = 0;
        float bestv = fabsf(cen[0] - d);
        for (int k = 1; k < PAIR_CEN_W; ++k) {
            float v = fabsf(cen[k] - d);
            if (v < bestv) { bestv = v; best = k; }   // first-min, matches jnp.argmin
        }
        const float* pots = ff_dist + (size_t)type * NBINS;
        float f = 0.5f * (pots[best] - pots[best + 2]);
        float inv = 1.f / fmaxf(d, 0.01f);
        fx += f * (cjx - c[3 * i])     * inv;
        fy += f * (cjy - c[3 * i + 1]) * inv;
        fz += f * (cjz - c[3 * i + 2]) * inv;
    }

    __shared__ float sx[256], sy[256], sz[256];
    int t = threadIdx.x;
    sx[t] = fx; sy[t] = fy; sz[t] = fz;
    __syncthreads();
    for (int s = 128; s > 0; s >>= 1) {
        if (t < s) { sx[t] += sx[t + s]; sy[t] += sy[t + s]; sz[t] += sz[t + s]; }
        __syncthreads();
    }
    if (t == 0) {
        float im = 1.f / masses[(size_t)bi * n + j];
        float* acc = wacc + (size_t)bi * n * 3;
        acc[3 * j]     = sx[0] * im;
        acc[3 * j + 1] = sy[0] * im;
        acc[3 * j + 2] = sz[0] * im;
    }
}

// ---------------- angle forces ----------------
__global__ void k_angle(const float* __restrict__ wc, const float* __restrict__ masses,
                        const float* __restrict__ ff_ang, const int* __restrict__ inters_ang,
                        float* __restrict__ wacc, int n_res, int n_ang_types, int b,
                        int step, const int* __restrict__ nsteps) {
    if (step >= *nsteps) return;
    int gid = blockIdx.x * blockDim.x + threadIdx.x;
    int per = 5 * n_res;
    if (gid >= b * per) return;
    int bi = gid / per;
    int w = gid % per;
    int ai = w / n_res;
    int r = w % n_res;

    const int I1[5] = {0, 1, 2, 0, 2};
    const int I2[5] = {1, 2, 0, 1, 1};
    const int I3[5] = {2, 0, 1, 3, 3};
    const int AX[5] = {0, 1, 2, 0, 0};
    int across = AX[ai];
    if (across != 0 && r >= n_res - 1) return;

    int n = 4 * n_res;
    const float* c = wc + (size_t)bi * n * 3;
    int r1 = r, r2 = r, r3 = r, potRow;
    if (across == 0)      { potRow = inters_ang[r]; }
    else if (across == 1) { r3 = r + 1; potRow = inters_ang[r]; }
    else                  { r2 = r + 1; r3 = r + 1; potRow = inters_ang[r + 1]; }

    int a1 = r1 * 4 + I1[ai], a2 = r2 * 4 + I2[ai], a3 = r3 * 4 + I3[ai];
    F3 A1 = loadc(c, a1), A2 = loadc(c, a2), A3 = loadc(c, a3);
    F3 ba = sub3(A1, A2);
    F3 bc = sub3(A3, A2);
    float ban = safe_norm3(ba), bcn = safe_norm3(bc);
    float ang = acosf(dot3(ba, bc) / (ban * bcn));

    double gap = (PI_D - PI_D / 3.0) / (double)NBINS;
    int best = 0; float bestv = 3.4e38f;
    for (int k = 0; k < NBINS - 2; ++k) {
        float cen = (float)(PI_D / 3.0 + ((double)k + 1.5) * gap);
        float v = fabsf(cen - ang);
        if (v < bestv) { bestv = v; best = k; }
    }
    const float* p = ff_ang + ((size_t)ai * n_ang_types + potRow) * ANG_POT_W;
    float f = 0.5f * (p[best] - p[best + 2]);

    F3 cr = cross3(ba, bc);
    F3 fa = scl3(normalize3(cross3(ba, cr)), f / ban);
    F3 fc = scl3(normalize3(cross3(neg3(bc), cr)), f / bcn);
    F3 fb = neg3(add3(fa, fc));

    const float* mass = masses + (size_t)bi * n;
    float* acc = wacc + (size_t)bi * n * 3;
    float m1 = 1.f / mass[a1], m2 = 1.f / mass[a2], m3 = 1.f / mass[a3];
    atomicAdd(&acc[3 * a1],     fa.x * m1);
    atomicAdd(&acc[3 * a1 + 1], fa.y * m1);
    atomicAdd(&acc[3 * a1 + 2], fa.z * m1);
    atomicAdd(&acc[3 * a2],     fb.x * m2);
    atomicAdd(&acc[3 * a2 + 1], fb.y * m2);
    atomicAdd(&acc[3 * a2 + 2], fb.z * m2);
    atomicAdd(&acc[3 * a3],     fc.x * m3);
    atomicAdd(&acc[3 * a3 + 1], fc.y * m3);
    atomicAdd(&acc[3 * a3 + 2], fc.z * m3);
}

// ---------------- dihedral forces ----------------
__global__ void k_dih(const float* __restrict__ wc, const float* __restrict__ masses,
                      const float* __restrict__ ff_dih, const int* __restrict__ inters_dih,
                      float* __restrict__ wacc, int n_res, int n_dih_types, int b,
                      int step, const int* __restrict__ nsteps) {
    if (step >= *nsteps) return;
    int gid = blockIdx.x * blockDim.x + threadIdx.x;
    int per = 5 * (n_res - 1);
    if (gid >= b * per) return;
    int bi = gid / per;
    int w = gid % per;
    int di = w / (n_res - 1);
    int r = w % (n_res - 1);

    const int I1[5] = {2, 0, 1, 2, 3};
    const int I2[5] = {0, 1, 2, 0, 1};
    const int I3[5] = {1, 2, 0, 1, 2};
    const int I4[5] = {2, 0, 1, 3, 0};
    const int AX[5] = {3, 1, 2, 3, 1};
    int across = AX[di];

    int n = 4 * n_res;
    const float* c = wc + (size_t)bi * n * 3;
    int rA = r, rB = r, rC = r, rD = r + 1, potRow;
    if (across == 1)      { potRow = inters_dih[r]; }
    else if (across == 2) { rC = r + 1; potRow = inters_dih[r + 1]; }
    else                  { rB = r + 1; rC = r + 1; potRow = inters_dih[r + 1]; }

    int a1 = rA * 4 + I1[di], a2 = rB * 4 + I2[di], a3 = rC * 4 + I3[di], a4 = rD * 4 + I4[di];
    F3 P1 = loadc(c, a1), P2 = loadc(c, a2), P3 = loadc(c, a3), P4 = loadc(c, a4);
    F3 ab = sub3(P2, P1);
    F3 bc = sub3(P3, P2);
    F3 cd = sub3(P4, P3);

    F3 c1 = cross3(ab, bc);
    F3 c2 = cross3(bc, cd);
    float bcn = safe_norm3(bc);
    F3 cc = cross3(c1, c2);
    float dih = atan2f(dot3(cc, bc) / bcn, dot3(c1, c2));

    double gap = 2.0 * PI_D / (double)NBINS;
    int best = 0; float bestv = 3.4e38f;
    for (int k = 0; k < NBINS; ++k) {
        float cen = (float)(-PI_D + ((double)k + 0.5) * gap);
        float v = fabsf(cen - dih);
        if (v < bestv) { bestv = v; best = k; }
    }
    const float* p = ff_dih + ((size_t)di * n_dih_types + potRow) * DIH_POT_W;
    float f = 0.5f * (p[best] - p[best + 2]);

    F3 fa = scl3(normalize3(neg3(c1)), f / safe_norm3(ab));
    F3 fd = scl3(normalize3(c2), f / safe_norm3(cd));
    float bsq = bcn * bcn;
    float t1 = dot3(ab, bc) / bsq;
    float t2 = dot3(cd, bc) / bsq;
    F3 fb = add3(add3(neg3(fa), scl3(fa, -t1)), scl3(fd, t2));
    F3 fcv = neg3(add3(add3(fa, fb), fd));

    const float* mass = masses + (size_t)bi * n;
    float* acc = wacc + (size_t)bi * n * 3;
    float m1 = 1.f / mass[a1], m2 = 1.f / mass[a2], m3 = 1.f / mass[a3], m4 = 1.f / mass[a4];
    atomicAdd(&acc[3 * a1],     fa.x * m1);
    atomicAdd(&acc[3 * a1 + 1], fa.y * m1);
    atomicAdd(&acc[3 * a1 + 2], fa.z * m1);
    atomicAdd(&acc[3 * a2],     fb.x * m2);
    atomicAdd(&acc[3 * a2 + 1], fb.y * m2);
    atomicAdd(&acc[3 * a2 + 2], fb.z * m2);
    atomicAdd(&acc[3 * a3],     fcv.x * m3);
    atomicAdd(&acc[3 * a3 + 1], fcv.y * m3);
    atomicAdd(&acc[3 * a3 + 2], fcv.z * m3);
    atomicAdd(&acc[3 * a4],     fd.x * m4);
    atomicAdd(&acc[3 * a4 + 1], fd.y * m4);
    atomicAdd(&acc[3 * a4 + 2], fd.z * m4);
}

// ---------------- velocity update + save acc_last ----------------
__global__ void k_vel(float* wv, const float* __restrict__ wacc, float* wal, int total,
                      int step, const int* __restrict__ nsteps) {
    if (step >= *nsteps) return;
    int i = blockIdx.x * blockDim.x + threadIdx.x;
    if (i < total) {
        float a = wacc[i];
        wv[i] += 0.5f * (wal[i] + a) * TIMESTEP;
        wal[i] = a;
    }
}

// ---------------- emit final coords ----------------
__global__ void k_out(const float* __restrict__ wc, float* out, int total) {
    int i = blockIdx.x * blockDim.x + threadIdx.x;
    if (i < total) out[i] = wc[i];
}

extern "C" void kernel_launch(void* const* d_in, const int* in_sizes, int n_in,
                              void* d_out, int out_size, void* d_ws, size_t ws_size,
                              hipStream_t stream) {
    (void)n_in; (void)out_size; (void)ws_size;
    const float* coords   = (const float*)d_in[0];
    const float* vels     = (const float*)d_in[1];
    const float* masses   = (const float*)d_in[2];
    const float* ff_dist  = (const float*)d_in[4];
    const float* ff_ang   = (const float*)d_in[5];
    const float* ff_dih   = (const float*)d_in[6];
    const float* bin_cen  = (const float*)d_in[7];
    const int* inters_flat = (const int*)d_in[8];
    const int* inters_ang  = (const int*)d_in[9];
    const int* inters_dih  = (const int*)d_in[10];
    const int* nsteps      = (const int*)d_in[11];

    int n_res = in_sizes[9];
    int n = 4 * n_res;
    int b = in_sizes[0] / (n * 3);
    int n_ang_types = in_sizes[5] / (5 * ANG_POT_W);
    int n_dih_types = in_sizes[6] / (5 * DIH_POT_W);

    // workspace layout (floats)
    float* ws    = (float*)d_ws;
    size_t bn3   = (size_t)b * n * 3;
    float* wc    = ws;
    float* wv    = wc + bn3;
    float* wacc  = wv + bn3;
    float* wal   = wacc + bn3;
    float* wn    = wal + bn3;
    float* wdist = wn + (size_t)b * n;

    int total3 = (int)bn3;
    int g3 = (total3 + 255) / 256;
    k_init<<<g3, 256, 0, stream>>>(coords, vels, wc, wv, wal, total3);

    int nt = (n + 15) / 16;
    int angWork = b * 5 * n_res;
    int dihWork = b * 5 * (n_res - 1);

    for (int s = 0; s < MAX_STEPS; ++s) {
        k_pos<<<g3, 256, 0, stream>>>(wc, wv, wal, total3, s, nsteps);
        k_norms<<<((b * n) + 255) / 256, 256, 0, stream>>>(wc, wn, b * n, s, nsteps);
        k_gram<<<dim3(nt, nt, b), 32, 0, stream>>>(wc, wn, wdist, n, s, nsteps);
        k_pair<<<dim3(n, b), 256, 0, stream>>>(wc, wdist, masses, ff_dist, bin_cen,
                                               inters_flat, wacc, n, s, nsteps);
        k_angle<<<(angWork + 127) / 128, 128, 0, stream>>>(wc, masses, ff_ang, inters_ang,
                                                           wacc, n_res, n_ang_types, b, s, nsteps);
        k_dih<<<(dihWork + 127) / 128, 128, 0, stream>>>(wc, masses, ff_dih, inters_dih,
                                                         wacc, n_res, n_dih_types, b, s, nsteps);
        k_vel<<<g3, 256, 0, stream>>>(wv, wacc, wal, total3, s, nsteps);
    }
    k_out<<<g3, 256, 0, stream>>>(wc, (float*)d_out, total3);
}